// FasttextSum_150_4449586119331
// MI455X (gfx1250) — compile-verified
//
#include <hip/hip_runtime.h>
#include <hip/hip_bf16.h>

// ---------------------------------------------------------------------------
// GNN pipeline on gfx1250 (MI455X):
//   L1: agg = scatter_add(features[src] -> dst); gcn1 = lrelu((agg+features)@W1+b1)
//   L2: agg = scatter_add(gcn1[src] -> dst);     gcn2 = lrelu((agg+gcn1)@W2+b2)
//   head: z = lrelu(gcn2[v1|v2] @ W3 + b3); out = l2norm(z)
// GEMMs use V_WMMA_F32_16X16X4_F32 (full f32 precision; K=364 = 91*4 = 13*28).
// B is staged in LDS as float2 (k-pair, n) so every B fragment is ONE aligned
// ds_load_b64 straight into an even VGPR pair (no v_mov marshalling).
// ---------------------------------------------------------------------------

typedef float v2f __attribute__((ext_vector_type(2)));
typedef float v8f __attribute__((ext_vector_type(8)));

#define GCN_NODES 50000
#define GCN_EDGES 200000
#define GCN_D     364
#define GCN_DOUT  150
#define GCN_PAIRS 8192

// GEMM tiling
#define BM 128            // block rows
#define BN 64             // block cols
#define KS 28             // K stage (364 = 13 * 28)
#define APITCH (KS + 2)   // LDS pitch for A stage (floats); even -> v2f aligned
#define BNP (BN + 1)      // LDS pitch for B stage (float2 elements)

__global__ void gcn_zero_kernel(float* __restrict__ p, size_t n) {
    size_t i = (size_t)blockIdx.x * blockDim.x + threadIdx.x;
    size_t stride = (size_t)gridDim.x * blockDim.x;
    for (; i < n; i += stride) p[i] = 0.0f;
}

// One block per edge: agg[dst[e]] += x[src[e]]  (364 floats, f32 atomics).
__global__ void gcn_scatter_add_kernel(const float* __restrict__ X,
                                       const int* __restrict__ src,
                                       const int* __restrict__ dst,
                                       float* __restrict__ AGG,
                                       int nE, int D) {
    int e = blockIdx.x;
    if (e >= nE) return;
    int s = src[e];
    int d = dst[e];
    const float* xs = X + (size_t)s * D;
    float*       ad = AGG + (size_t)d * D;
    for (int f = threadIdx.x; f < D; f += blockDim.x)
        atomicAdd(&ad[f], xs[f]);
}

// C[m,n] = lrelu( sum_k A'[m,k]*B[k,n] + bias[n] )
//   A'[m,:] = A[node(m),:] (+ AGG[m,:] if AGG != null)
//   node(m) = m, or idx1/idx2 gather for the head GEMM.
// Requires K % KS == 0 (true: 364 = 13*28).
__global__ __launch_bounds__(256)
void gcn_gemm_wmma_f32(const float* __restrict__ A,
                       const float* __restrict__ AGG,   // optional residual (nullptr for head)
                       const int*   __restrict__ idx1,  // optional gather, first half
                       const int*   __restrict__ idx2,  // optional gather, second half
                       const float* __restrict__ B,     // [K x N] row-major
                       const float* __restrict__ bias,  // [N]
                       float* __restrict__ C,           // [M x ldc]
                       int M, int N, int K, int ldc) {
    __shared__ float As[BM * APITCH];
    __shared__ v2f   Bs2[(KS / 2) * BNP];   // [k-pair][n] = {B[2p][n], B[2p+1][n]}

    const int tid  = threadIdx.x;
    const int lane = tid & 31;          // wave32
    const int wave = tid >> 5;          // 0..7
    const int wm   = wave >> 1;         // 0..3 : wave row (M)
    const int wn   = wave & 1;          // 0..1 : wave col (N)
    const int half = lane >> 4;         // 0 or 1 (lane half)
    const int l16  = lane & 15;

    const int block_m0 = blockIdx.y * BM;
    const int block_n0 = blockIdx.x * BN;
    const int Mhalf = M >> 1;

    v8f acc[2][2];
#pragma unroll
    for (int i = 0; i < 2; ++i)
#pragma unroll
        for (int j = 0; j < 2; ++j)
#pragma unroll
            for (int r = 0; r < 8; ++r) acc[i][j][r] = 0.0f;

    for (int k0 = 0; k0 < K; k0 += KS) {
        // Speculative prefetch of next B stage -> global_prefetch_b8
        if (k0 + KS < K) {
            int pc = block_n0 + (tid & 63);
            if (pc < N)
                __builtin_prefetch(&B[(size_t)(k0 + KS) * N + pc], 0, 0);
        }

        // Stage A block [BM x KS] into LDS (with gather / residual fused).
        for (int i = tid; i < BM * KS; i += 256) {
            int r = i / KS, c = i - r * KS;
            int gr = block_m0 + r;
            float v = 0.0f;
            if (gr < M) {
                int node = gr;
                if (idx1) node = (gr < Mhalf) ? idx1[gr] : idx2[gr - Mhalf];
                v = A[(size_t)node * K + (k0 + c)];
                if (AGG) v += AGG[(size_t)gr * K + (k0 + c)];
            }
            As[r * APITCH + c] = v;
        }
        // Stage B block [KS x BN] into LDS, fragment-ready float2 layout:
        // Bs2[p*BNP + n] = { B[k0+2p][n0+n], B[k0+2p+1][n0+n] }  (zero-pad n >= N).
        for (int i = tid; i < KS * BN; i += 256) {
            int r = i >> 6, c = i & 63;          // BN == 64
            int gc = block_n0 + c;
            float v = (gc < N) ? B[(size_t)(k0 + r) * N + gc] : 0.0f;
            ((float*)&Bs2[(r >> 1) * BNP + c])[r & 1] = v;
        }
        __syncthreads();

        const int am0 = wm * 32;   // wave's 2 M-tiles: am0, am0+16
        const int bn0 = wn * 32;   // wave's 2 N-tiles: bn0, bn0+16

#pragma unroll
        for (int kk = 0; kk < KS; kk += 4) {
            // A frag 16x4: lane l<16 holds A[l][kk..kk+1]; lane l+16 holds A[l][kk+2..kk+3]
            v2f afrag[2];
#pragma unroll
            for (int t = 0; t < 2; ++t)
                afrag[t] = *(const v2f*)&As[(am0 + t * 16 + l16) * APITCH + kk + 2 * half];
            // B frag 4x16: lane l<16 holds B[kk..kk+1][n0+l]; lane l+16 holds B[kk+2..kk+3][n0+l]
            // One aligned ds_load_b64 per tile from the float2-staged buffer.
            v2f bfrag[2];
#pragma unroll
            for (int t = 0; t < 2; ++t)
                bfrag[t] = Bs2[((kk >> 1) + half) * BNP + bn0 + t * 16 + l16];
#pragma unroll
            for (int i = 0; i < 2; ++i)
#pragma unroll
                for (int j = 0; j < 2; ++j)
                    acc[i][j] = __builtin_amdgcn_wmma_f32_16x16x4_f32(
                        /*neg_a=*/false, afrag[i],
                        /*neg_b=*/false, bfrag[j],
                        /*c_mod=*/(short)0, acc[i][j],
                        /*reuse_a=*/false, /*reuse_b=*/false);
        }
        __syncthreads();
    }

    // Epilogue: bias + leaky-ReLU.  C/D layout: VGPR r -> row r (lanes 0-15)
    // or row 8+r (lanes 16-31); col = n0 + (lane & 15).
#pragma unroll
    for (int i = 0; i < 2; ++i) {
        int mt = block_m0 + wm * 32 + i * 16;
#pragma unroll
        for (int j = 0; j < 2; ++j) {
            int col = block_n0 + wn * 32 + j * 16 + l16;
#pragma unroll
            for (int r = 0; r < 8; ++r) {
                int row = mt + half * 8 + r;
                if (row < M && col < N) {
                    float v = acc[i][j][r] + bias[col];
                    v = (v > 0.0f) ? v : v * 0.01f;
                    C[(size_t)row * ldc + col] = v;
                }
            }
        }
    }
}

// One wave per row: out[r,:] = Z[r,:] / max(||Z[r,:]||, 1e-12)
__global__ void gcn_l2norm_kernel(const float* __restrict__ Z,
                                  float* __restrict__ out,
                                  int rows, int cols) {
    int w = (int)((blockIdx.x * (size_t)blockDim.x + threadIdx.x) >> 5);
    int lane = threadIdx.x & 31;
    if (w >= rows) return;
    const float* zr = Z + (size_t)w * cols;
    float s = 0.0f;
    for (int c = lane; c < cols; c += 32) { float v = zr[c]; s += v * v; }
#pragma unroll
    for (int off = 16; off > 0; off >>= 1) s += __shfl_xor(s, off, 32);
    float scale = 1.0f / fmaxf(sqrtf(s), 1e-12f);
    float* o = out + (size_t)w * cols;
    for (int c = lane; c < cols; c += 32) o[c] = zr[c] * scale;
}

extern "C" void kernel_launch(void* const* d_in, const int* in_sizes, int n_in,
                              void* d_out, int out_size, void* d_ws, size_t ws_size,
                              hipStream_t stream) {
    const float* features = (const float*)d_in[0];
    const int*   src      = (const int*)d_in[1];
    const int*   dst      = (const int*)d_in[2];
    const int*   v1       = (const int*)d_in[3];
    const int*   v2       = (const int*)d_in[4];
    const float* W1       = (const float*)d_in[5];
    const float* b1       = (const float*)d_in[6];
    const float* W2       = (const float*)d_in[7];
    const float* b2       = (const float*)d_in[8];
    const float* W3       = (const float*)d_in[9];
    const float* b3       = (const float*)d_in[10];
    float* out = (float*)d_out;

    const size_t nodeFeats = (size_t)GCN_NODES * GCN_D;  // 18.2M floats
    float* AGG = (float*)d_ws;          // [NODES x D]
    float* G1  = AGG + nodeFeats;       // [NODES x D]
    float* G2  = G1 + nodeFeats;        // [NODES x D]
    float* Z   = AGG;                   // head output reuses AGG (last read: layer-2 GEMM)

    const int M2 = 2 * GCN_PAIRS;       // 16384 gathered rows

    dim3 gemm_grid_l((GCN_D + BN - 1) / BN, (GCN_NODES + BM - 1) / BM);
    dim3 gemm_grid_h((GCN_DOUT + BN - 1) / BN, (M2 + BM - 1) / BM);

    // ---- Layer 1 ----
    gcn_zero_kernel<<<2048, 256, 0, stream>>>(AGG, nodeFeats);
    gcn_scatter_add_kernel<<<GCN_EDGES, 384, 0, stream>>>(features, src, dst, AGG,
                                                          GCN_EDGES, GCN_D);
    gcn_gemm_wmma_f32<<<gemm_grid_l, 256, 0, stream>>>(
        features, AGG, nullptr, nullptr, W1, b1, G1,
        GCN_NODES, GCN_D, GCN_D, GCN_D);

    // ---- Layer 2 ----
    gcn_zero_kernel<<<2048, 256, 0, stream>>>(AGG, nodeFeats);
    gcn_scatter_add_kernel<<<GCN_EDGES, 384, 0, stream>>>(G1, src, dst, AGG,
                                                          GCN_EDGES, GCN_D);
    gcn_gemm_wmma_f32<<<gemm_grid_l, 256, 0, stream>>>(
        G1, AGG, nullptr, nullptr, W2, b2, G2,
        GCN_NODES, GCN_D, GCN_D, GCN_D);

    // ---- Head: gather(v1|v2) @ W3 + b3, lrelu ----
    gcn_gemm_wmma_f32<<<gemm_grid_h, 256, 0, stream>>>(
        G2, nullptr, v1, v2, W3, b3, Z,
        M2, GCN_DOUT, GCN_D, GCN_DOUT);

    // ---- Row-wise L2 normalize into d_out (z1 then z2, already contiguous) ----
    gcn_l2norm_kernel<<<(M2 + 7) / 8, 256, 0, stream>>>(Z, out, M2, GCN_DOUT);
}